// GCNLayer_9715216023647
// MI455X (gfx1250) — compile-verified
//
#include <hip/hip_runtime.h>
#include <math.h>

typedef float v2f __attribute__((ext_vector_type(2)));
typedef float v8f __attribute__((ext_vector_type(8)));

#define EPS      1e-15f
#define PROJ_EPS 1e-5f

__device__ __forceinline__ float wave_reduce_add(float v) {
    v += __shfl_xor(v, 1, 32);
    v += __shfl_xor(v, 2, 32);
    v += __shfl_xor(v, 4, 32);
    v += __shfl_xor(v, 8, 32);
    v += __shfl_xor(v, 16, 32);
    return v;
}

// ---------------------------------------------------------------------------
// Kernel 1: fused log_map_zero + dense GEMM (tangent @ W) using f32 WMMA.
// 32 rows per block, 8 waves; wave w owns output columns [16w,16w+16) and
// both 16-row tiles. K accumulated in steps of 4 via V_WMMA_F32_16X16X4_F32.
// ---------------------------------------------------------------------------
__global__ __launch_bounds__(256) void logmap_gemm_kernel(
    const float* __restrict__ x, const float* __restrict__ W,
    float* __restrict__ mapped, int N)
{
    __shared__ float xs[32][132];   // padded: rows hit distinct banks
    __shared__ float sc[32];

    const int tid     = threadIdx.x;
    const int rowbase = blockIdx.x * 32;

    // Load 32x128 tile of x as float4 (coalesced).
    for (int i = 0; i < 4; ++i) {
        int f   = tid + i * 256;          // float4 index, 0..1023
        int r   = f >> 5;                 // row in tile
        int c4  = f & 31;                 // float4 column
        int gr  = rowbase + r;
        gr      = gr < N ? gr : N - 1;    // clamp (dup rows, stores guarded)
        float4 v = ((const float4*)x)[(size_t)gr * 32 + c4];
        *(float4*)&xs[r][c4 * 4] = v;
    }
    __syncthreads();

    const int wave = tid >> 5;
    const int lane = tid & 31;

    // Per-row squared norm: wave handles rows 4w..4w+3, 8 lanes per row.
    {
        int subrow = lane >> 3;
        int sl     = lane & 7;
        int r      = wave * 4 + subrow;
        float s = 0.f;
        #pragma unroll
        for (int c = 0; c < 16; ++c) {
            float v = xs[r][sl * 16 + c];
            s += v * v;
        }
        s += __shfl_xor(s, 1, 8);
        s += __shfl_xor(s, 2, 8);
        s += __shfl_xor(s, 4, 8);
        float n = fmaxf(sqrtf(s), EPS);
        float z = fminf(n, 1.0f - 1e-7f);     // atanh clip (n >= 0)
        if (sl == 0) sc[r] = atanhf(z) / n;   // log_map_zero scale
    }
    __syncthreads();

    // Apply log-map scale in LDS.
    for (int i = 0; i < 4; ++i) {
        int f  = tid + i * 256;
        int r  = f >> 5;
        int c4 = f & 31;
        float s  = sc[r];
        float4 v = *(float4*)&xs[r][c4 * 4];
        v.x *= s; v.y *= s; v.z *= s; v.w *= s;
        *(float4*)&xs[r][c4 * 4] = v;
    }
    __syncthreads();

    // WMMA: A 16x4 f32 layout -> lane {0..15}=M, hi half holds K+2..K+3.
    const int sl    = lane & 15;
    const int hi    = lane >> 4;          // 0 or 1
    const int cbase = wave * 16;
    v8f acc0 = {};
    v8f acc1 = {};
    for (int k = 0; k < 128; k += 4) {
        const int kk = k + 2 * hi;
        v2f b;
        b.x = W[kk * 128 + cbase + sl];
        b.y = W[(kk + 1) * 128 + cbase + sl];
        v2f a0 = *(const v2f*)&xs[sl][kk];
        v2f a1 = *(const v2f*)&xs[16 + sl][kk];
        acc0 = __builtin_amdgcn_wmma_f32_16x16x4_f32(
            false, a0, false, b, (short)0, acc0, false, false);
        acc1 = __builtin_amdgcn_wmma_f32_16x16x4_f32(
            false, a1, false, b, (short)0, acc1, false, false);
    }

    // C/D layout: VGPR r -> M=r (lanes 0-15), M=r+8 (lanes 16-31); N=sl.
    #pragma unroll
    for (int r = 0; r < 8; ++r) {
        int row0 = rowbase + r + 8 * hi;
        int row1 = rowbase + 16 + r + 8 * hi;
        if (row0 < N) mapped[(size_t)row0 * 128 + cbase + sl] = acc0[r];
        if (row1 < N) mapped[(size_t)row1 * 128 + cbase + sl] = acc1[r];
    }
}

// ---------------------------------------------------------------------------
// Kernel 2: edge scatter-add. One wave per edge; float4 gather of mapped[col],
// 4 hardware f32 atomics per lane into agg[row] (L2-resident destination).
// ---------------------------------------------------------------------------
__global__ __launch_bounds__(256) void aggregate_kernel(
    const float* __restrict__ mapped, const float* __restrict__ val,
    const int* __restrict__ row, const int* __restrict__ col,
    float* __restrict__ agg, int E)
{
    const int e = blockIdx.x * 8 + (threadIdx.x >> 5);
    if (e >= E) return;
    const int lane = threadIdx.x & 31;
    const int r = row[e];
    const int c = col[e];
    const float v = val[e];
    float4 m = ((const float4*)mapped)[(size_t)c * 32 + lane];
    float* dst = &agg[(size_t)r * 128 + lane * 4];
    unsafeAtomicAdd(dst + 0, v * m.x);
    unsafeAtomicAdd(dst + 1, v * m.y);
    unsafeAtomicAdd(dst + 2, v * m.z);
    unsafeAtomicAdd(dst + 3, v * m.w);
}

// ---------------------------------------------------------------------------
// Kernel 3: in-place finalize: h = proj(exp0(agg)); b = proj(exp0(bias));
// out = proj(mobius_add(h, b)). One wave per row (4 floats per lane).
// ---------------------------------------------------------------------------
__global__ __launch_bounds__(256) void finalize_kernel(
    float* __restrict__ out, const float* __restrict__ bias, int N)
{
    const int wave = threadIdx.x >> 5;
    const int lane = threadIdx.x & 31;
    const int rowi = blockIdx.x * 8 + wave;

    // --- bias branch: b = proj(exp_map_zero(bias))  (redundant per wave) ---
    float4 bv = ((const float4*)bias)[lane];
    float nb2 = wave_reduce_add(bv.x * bv.x + bv.y * bv.y +
                                bv.z * bv.z + bv.w * bv.w);
    float nb = fmaxf(sqrtf(nb2), EPS);
    float tb = tanhf(fminf(nb, 15.f));
    float sb = (tb / nb) * fminf(1.f, (1.f - PROJ_EPS) / fmaxf(tb, EPS));
    float4 b4 = { bv.x * sb, bv.y * sb, bv.z * sb, bv.w * sb };

    if (rowi >= N) return;

    // --- h = proj(exp_map_zero(agg_row)) ---
    float4 a4 = ((const float4*)out)[(size_t)rowi * 32 + lane];
    float n2 = wave_reduce_add(a4.x * a4.x + a4.y * a4.y +
                               a4.z * a4.z + a4.w * a4.w);
    float n  = fmaxf(sqrtf(n2), EPS);
    float t  = tanhf(fminf(n, 15.f));
    float hs = (t / n) * fminf(1.f, (1.f - PROJ_EPS) / fmaxf(t, EPS));
    float4 h4 = { a4.x * hs, a4.y * hs, a4.z * hs, a4.w * hs };

    // --- mobius_addition(h, b) ---
    float x2 = wave_reduce_add(h4.x * h4.x + h4.y * h4.y +
                               h4.z * h4.z + h4.w * h4.w);
    float y2 = wave_reduce_add(b4.x * b4.x + b4.y * b4.y +
                               b4.z * b4.z + b4.w * b4.w);
    float xy = wave_reduce_add(h4.x * b4.x + h4.y * b4.y +
                               h4.z * b4.z + h4.w * b4.w);
    float ch = 1.f + 2.f * xy + y2;
    float cb = 1.f - x2;
    float den = fmaxf(1.f + 2.f * xy + x2 * y2, EPS);
    float inv = 1.f / den;
    float4 o4 = { (ch * h4.x + cb * b4.x) * inv,
                  (ch * h4.y + cb * b4.y) * inv,
                  (ch * h4.z + cb * b4.z) * inv,
                  (ch * h4.w + cb * b4.w) * inv };

    // --- final projection ---
    float on2 = wave_reduce_add(o4.x * o4.x + o4.y * o4.y +
                                o4.z * o4.z + o4.w * o4.w);
    float os = fminf(1.f, (1.f - PROJ_EPS) / fmaxf(sqrtf(on2), EPS));
    float4 r4 = { o4.x * os, o4.y * os, o4.z * os, o4.w * os };
    ((float4*)out)[(size_t)rowi * 32 + lane] = r4;
}

extern "C" void kernel_launch(void* const* d_in, const int* in_sizes, int n_in,
                              void* d_out, int out_size, void* d_ws, size_t ws_size,
                              hipStream_t stream) {
    const float* x    = (const float*)d_in[0];   // [N,128]
    const float* W    = (const float*)d_in[1];   // [128,128]
    const float* bias = (const float*)d_in[2];   // [1,128]
    const float* val  = (const float*)d_in[3];   // [E]
    const int*   row  = (const int*)d_in[4];     // [E]
    const int*   col  = (const int*)d_in[5];     // [E]

    const int N = in_sizes[0] / 128;
    const int E = in_sizes[3];

    float* out    = (float*)d_out;
    float* mapped = (float*)d_ws;                // [N,128] scratch, 51.2 MB

    // Zero the atomic accumulation target (d_out). Async memset is capturable.
    hipMemsetAsync(d_out, 0, (size_t)out_size * sizeof(float), stream);

    logmap_gemm_kernel<<<(N + 31) / 32, 256, 0, stream>>>(x, W, mapped, N);
    aggregate_kernel<<<(E + 7) / 8, 256, 0, stream>>>(mapped, val, row, col, out, E);
    finalize_kernel<<<(N + 7) / 8, 256, 0, stream>>>(out, bias, N);
}